// CRGNN_57363583205979
// MI455X (gfx1250) — compile-verified
//
#include <hip/hip_runtime.h>
#include <hip/hip_bf16.h>

typedef __attribute__((ext_vector_type(2))) float v2f;
typedef __attribute__((ext_vector_type(8))) float v8f;
typedef int v4i_vs __attribute__((vector_size(16)));   // matches builtin param type

#define NNODE 512
#define DHID  256
#define NW    16      // 512 bits = 16 u32 words

// ---------------------------------------------------------------------------
// Optional CDNA5 async global->LDS copy (ASYNCcnt-tracked DMA path)
// ---------------------------------------------------------------------------
#if defined(__has_builtin)
#if __has_builtin(__builtin_amdgcn_global_load_async_to_lds_b128)
#define HAVE_ASYNC_LDS 1
#endif
#if __has_builtin(__builtin_amdgcn_s_wait_asynccnt)
#define HAVE_WAIT_ASYNC 1
#endif
#endif

__device__ __forceinline__ void stage_b128(const float* src, float* dstLds) {
#if defined(HAVE_ASYNC_LDS)
    __builtin_amdgcn_global_load_async_to_lds_b128(
        (__attribute__((address_space(1))) v4i_vs*)src,
        (__attribute__((address_space(3))) v4i_vs*)dstLds, 0, 0);
#else
    *(float4*)dstLds = *(const float4*)src;
#endif
}

__device__ __forceinline__ void stage_fence() {
#if defined(HAVE_ASYNC_LDS)
#if defined(HAVE_WAIT_ASYNC)
    __builtin_amdgcn_s_wait_asynccnt(0);
#else
    asm volatile("s_wait_asynccnt 0" ::: "memory");
#endif
#endif
    __syncthreads();
}

// ---------------------------------------------------------------------------
// Wave helpers (wave32)
// ---------------------------------------------------------------------------
__device__ __forceinline__ float wave_sum_f32(float v) {
#pragma unroll
    for (int o = 16; o > 0; o >>= 1) v += __shfl_xor(v, o, 32);
    return v;
}

// ---------------------------------------------------------------------------
// 1) adjacency -> bitset rows (512 x 16 u32)
// ---------------------------------------------------------------------------
__global__ void crgnn_adj_to_bits(const float* __restrict__ adj,
                                  unsigned int* __restrict__ bits) {
    const int v = blockIdx.x;
    const int lane = threadIdx.x;
#pragma unroll
    for (int w = 0; w < NW; ++w) {
        float a = adj[v * NNODE + w * 32 + lane];
        unsigned long long m = __ballot(a != 0.0f);
        if (lane == 0) bits[v * NW + w] = (unsigned int)m;
    }
}

// ---------------------------------------------------------------------------
// 2) CR features (cols 0..3) + degree features (cols 12..14), one wave / node
// ---------------------------------------------------------------------------
__global__ void crgnn_cr_features(const unsigned int* __restrict__ bits,
                                  float* __restrict__ feats) {
    const int v = blockIdx.x;
    const int lane = threadIdx.x;

    unsigned int m[NW];
    int degi = 0;
#pragma unroll
    for (int w = 0; w < NW; ++w) {
        unsigned int r = bits[v * NW + w];
        degi += __popc(r);
        if (w == (v >> 5)) r |= (1u << (v & 31));   // M = A + I
        m[w] = r;
    }

    // E, P2
    float E2 = 0.0f, P2a = 0.0f;
#pragma unroll
    for (int j = 0; j < NW; ++j) {
        if (m[j] & (1u << lane)) {
            const int b = lane + 32 * j;
            int d = 0;
#pragma unroll
            for (int w = 0; w < NW; ++w) d += __popc(m[w] & bits[b * NW + w]);
            const float df = (float)d;
            E2 += df;
            P2a += df * (df - 1.0f);
        }
    }

    // t6_v = sum_{a,c in S_v, A_ac=1} popc(m & row_a & row_c)
    float t6 = 0.0f;
#pragma unroll
    for (int aw = 0; aw < NW; ++aw) {
        unsigned int mw = m[aw];
        while (mw) {
            const int a = aw * 32 + (__ffs(mw) - 1);
            mw &= mw - 1;
            unsigned int ra[NW];
#pragma unroll
            for (int w = 0; w < NW; ++w) ra[w] = m[w] & bits[a * NW + w];
#pragma unroll
            for (int j = 0; j < NW; ++j) {
                if (ra[j] & (1u << lane)) {          // c in S_v and c ~ a
                    const int c = lane + 32 * j;
                    int p = 0;
#pragma unroll
                    for (int w = 0; w < NW; ++w) p += __popc(ra[w] & bits[c * NW + w]);
                    t6 += (float)p;
                }
            }
        }
    }

    E2  = wave_sum_f32(E2);
    P2a = wave_sum_f32(P2a);
    t6  = wave_sum_f32(t6);

    if (lane == 0) {
        const float deg = (float)degi;
        const float s = deg + 1.0f;
        const float E = 0.5f * E2;
        const float P2 = 0.5f * P2a;
        const float t3 = t6 / 6.0f;
        const float T = s * (s - 1.0f) * (s - 2.0f) / 6.0f;
        float n3 = t3;
        float n2 = P2 - 3.0f * t3;
        float n1 = E * (s - 2.0f) - 2.0f * n2 - 3.0f * n3;
        float n0 = T - n1 - n2 - n3;
        if (s < 3.0f) { n0 = n1 = n2 = n3 = 0.0f; }
        const float den = n0 + n1 + n2 + n3 + 1e-10f;
        float* f = feats + v * 15;
        f[0] = n0 / den; f[1] = n1 / den; f[2] = n2 / den; f[3] = n3 / den;
        f[12] = deg; f[13] = deg * deg; f[14] = deg;   // diag(A@A) == deg
    }
}

// ---------------------------------------------------------------------------
// 3) fp32 WMMA GEMM: Out[M,N] = A[M,K] * B(+flags) with flexible C-init
//    block: 128 threads (4 waves), 64x64 tile, LDS-staged K step of 16 via
//    128-bit (optionally async) copies. Each wave: 2x2 grid of 16x16
//    accumulators chained through V_WMMA_F32_16X16X4_F32.
//    LDS layouts: As[row][k]    pitch 20  (b128-aligned, conflict-free)
//                 B  [k][n]     pitch 68  (non-transposed B)
//                 Bt [n][k]     pitch 20  (transposed B: contiguous k loads)
// ---------------------------------------------------------------------------
#define TM 64
#define TN 64
#define TK 16
#define APITCH 20
#define BPITCH 68

__global__ __launch_bounds__(128)
void crgnn_gemm_wmma(const float* A, int lda,
                     const float* B, int ldb, int transB,
                     const float* Cinit, int ldc,
                     const float* bias,
                     const float* epsArr, int epsIdx,
                     float* Out, int ldo,
                     int K, int relu) {
    __shared__ float As[TM * APITCH];     // 64 x 20
    __shared__ float Bsh[TM * APITCH];    // max(16*68, 64*20) = 1280 floats

    const int tid  = threadIdx.x;
    const int lane = tid & 31;
    const int wave = tid >> 5;
    const int hi   = lane >> 4;      // 0/1 half of the wave
    const int lo   = lane & 15;

    const int rowBase = blockIdx.y * TM;
    const int colBase = blockIdx.x * TN;
    const int wrow = (wave >> 1) * 32;
    const int wcol = (wave & 1) * 32;

    float scale = 0.0f;
    if (Cinit && epsArr) scale = 1.0f + epsArr[epsIdx];

    v8f acc[2][2];
#pragma unroll
    for (int rt = 0; rt < 2; ++rt)
#pragma unroll
        for (int ct = 0; ct < 2; ++ct) {
            const int col = colBase + wcol + ct * 16 + lo;
            const float binit = bias ? bias[col] : 0.0f;
#pragma unroll
            for (int j = 0; j < 8; ++j) {
                const int row = rowBase + wrow + rt * 16 + j + 8 * hi;
                float cv = binit;
                if (Cinit) cv = scale * Cinit[row * ldc + col];
                acc[rt][ct][j] = cv;
            }
        }

    for (int k0 = 0; k0 < K; k0 += TK) {
        // ---- stage A tile (64x16): 256 b128 transfers, 2 per thread
#pragma unroll
        for (int r = 0; r < 2; ++r) {
            const int idx = tid + r * 128;
            const int row = idx >> 2;
            const int q   = (idx & 3) * 4;
            stage_b128(A + (rowBase + row) * lda + k0 + q, &As[row * APITCH + q]);
        }
        // ---- stage B tile (16x64): 256 b128 transfers, 2 per thread
        if (!transB) {
#pragma unroll
            for (int r = 0; r < 2; ++r) {
                const int idx = tid + r * 128;
                const int k   = idx >> 4;
                const int nq  = (idx & 15) * 4;
                stage_b128(B + (k0 + k) * ldb + colBase + nq,
                           &Bsh[k * BPITCH + nq]);
            }
        } else {
#pragma unroll
            for (int r = 0; r < 2; ++r) {
                const int idx = tid + r * 128;
                const int n   = idx >> 2;
                const int kq  = (idx & 3) * 4;
                stage_b128(B + (colBase + n) * ldb + k0 + kq,
                           &Bsh[n * APITCH + kq]);
            }
        }
        stage_fence();

#pragma unroll
        for (int kk = 0; kk < TK; kk += 4) {
            v2f af[2], bf[2];
#pragma unroll
            for (int rt = 0; rt < 2; ++rt) {
                const int r = wrow + rt * 16 + lo;
                af[rt].x = As[r * APITCH + kk + 2 * hi + 0];  // lane<16: K=0,1 ; lane>=16: K=2,3
                af[rt].y = As[r * APITCH + kk + 2 * hi + 1];
            }
#pragma unroll
            for (int ct = 0; ct < 2; ++ct) {
                const int c = wcol + ct * 16 + lo;
                if (!transB) {                                // B 4x16: v0 rows {0,2}, v1 rows {1,3}
                    bf[ct].x = Bsh[(kk + 2 * hi + 0) * BPITCH + c];
                    bf[ct].y = Bsh[(kk + 2 * hi + 1) * BPITCH + c];
                } else {
                    bf[ct].x = Bsh[c * APITCH + kk + 2 * hi + 0];
                    bf[ct].y = Bsh[c * APITCH + kk + 2 * hi + 1];
                }
            }
#pragma unroll
            for (int rt = 0; rt < 2; ++rt)
#pragma unroll
                for (int ct = 0; ct < 2; ++ct)
                    acc[rt][ct] = __builtin_amdgcn_wmma_f32_16x16x4_f32(
                        false, af[rt], false, bf[ct], (short)0, acc[rt][ct],
                        false, false);
        }
        __syncthreads();
    }

#pragma unroll
    for (int rt = 0; rt < 2; ++rt)
#pragma unroll
        for (int ct = 0; ct < 2; ++ct) {
            const int col = colBase + wcol + ct * 16 + lo;
#pragma unroll
            for (int j = 0; j < 8; ++j) {
                const int row = rowBase + wrow + rt * 16 + j + 8 * hi;
                float vv = acc[rt][ct][j];
                if (relu) vv = vv > 0.0f ? vv : 0.0f;
                Out[row * ldo + col] = vv;
            }
        }
}

// ---------------------------------------------------------------------------
// 4) per-row top-8 of A^4 -> feats cols 4..11 (one wave per row)
// ---------------------------------------------------------------------------
__global__ void crgnn_top8(const float* __restrict__ A4,
                           float* __restrict__ feats) {
    const int v = blockIdx.x;
    const int lane = threadIdx.x;
    float val[16];
#pragma unroll
    for (int j = 0; j < 16; ++j) val[j] = A4[v * NNODE + lane + 32 * j];
    unsigned int taken = 0;
    for (int r = 0; r < 8; ++r) {
        float bv = -1.0f;   // A^4 entries are >= 0
        int bj = 0;
#pragma unroll
        for (int j = 0; j < 16; ++j)
            if (!((taken >> j) & 1u) && val[j] > bv) { bv = val[j]; bj = j; }
        int bl = lane;
#pragma unroll
        for (int o = 16; o > 0; o >>= 1) {
            const float ov = __shfl_xor(bv, o, 32);
            const int   ol = __shfl_xor(bl, o, 32);
            if (ov > bv || (ov == bv && ol < bl)) { bv = ov; bl = ol; }
        }
        if (lane == bl) taken |= (1u << bj);
        if (lane == 0) feats[v * 15 + 4 + r] = bv;
    }
}

// ---------------------------------------------------------------------------
// 5) embed: x[m][n] = b[n] + sum_k feat[m][k] * W[n][k]   (K = 15)
// ---------------------------------------------------------------------------
__global__ void crgnn_embed(const float* __restrict__ feats,
                            const float* __restrict__ W,
                            const float* __restrict__ b,
                            float* __restrict__ x) {
    const int m = blockIdx.x;
    const int n = threadIdx.x;
    float s = b[n];
#pragma unroll
    for (int k = 0; k < 15; ++k) s += feats[m * 15 + k] * W[n * 15 + k];
    x[m * DHID + n] = s;
}

// ---------------------------------------------------------------------------
// 6) column-sum reduction: out[n] = sum_m x[m][n]
// ---------------------------------------------------------------------------
__global__ void crgnn_colsum(const float* __restrict__ x,
                             float* __restrict__ out) {
    const int n = threadIdx.x;
    float s = 0.0f;
    for (int m = 0; m < NNODE; ++m) s += x[m * DHID + n];
    out[n] = s;
}

// ---------------------------------------------------------------------------
// Launch
// ---------------------------------------------------------------------------
extern "C" void kernel_launch(void* const* d_in, const int* in_sizes, int n_in,
                              void* d_out, int out_size, void* d_ws, size_t ws_size,
                              hipStream_t stream) {
    (void)in_sizes; (void)n_in; (void)out_size; (void)ws_size;
    const float* adj     = (const float*)d_in[0];   // 512*512
    const float* W_embed = (const float*)d_in[1];   // 256*15
    const float* b_embed = (const float*)d_in[2];   // 256
    const float* W1      = (const float*)d_in[3];   // 3*256*256
    const float* b1      = (const float*)d_in[4];   // 3*256
    const float* W2      = (const float*)d_in[5];   // 3*256*256
    const float* b2      = (const float*)d_in[6];   // 3*256
    const float* eps     = (const float*)d_in[7];   // 3
    float* out = (float*)d_out;

    char* ws = (char*)d_ws;
    unsigned int* bits = (unsigned int*)(ws + 0);          // 32 KB
    float* feats = (float*)(ws + 32768);                   // 30 KB (pad 32 KB)
    float* A2    = (float*)(ws + 65536);                   // 1 MB
    float* A4    = (float*)(ws + 65536 + 1048576);         // 1 MB
    float* xA    = (float*)(ws + 2162688);                 // 512 KB
    float* xB    = (float*)(ws + 2686976);                 // 512 KB
    float* agg   = (float*)(ws + 3211264);                 // 512 KB
    float* hbuf  = (float*)(ws + 3735552);                 // 512 KB

    crgnn_adj_to_bits<<<NNODE, 32, 0, stream>>>(adj, bits);
    crgnn_cr_features<<<NNODE, 32, 0, stream>>>(bits, feats);

    // A2 = adj @ adj ; A4 = A2 @ A2   (exact integer fp32 via WMMA)
    crgnn_gemm_wmma<<<dim3(8, 8), 128, 0, stream>>>(
        adj, NNODE, adj, NNODE, 0, nullptr, 0, nullptr, nullptr, 0,
        A2, NNODE, NNODE, 0);
    crgnn_gemm_wmma<<<dim3(8, 8), 128, 0, stream>>>(
        A2, NNODE, A2, NNODE, 0, nullptr, 0, nullptr, nullptr, 0,
        A4, NNODE, NNODE, 0);

    crgnn_top8<<<NNODE, 32, 0, stream>>>(A4, feats);
    crgnn_embed<<<NNODE, DHID, 0, stream>>>(feats, W_embed, b_embed, xA);

    float* xc = xA;
    float* xn = xB;
    for (int l = 0; l < 3; ++l) {
        // agg = (1+eps[l]) * x + adj @ x
        crgnn_gemm_wmma<<<dim3(4, 8), 128, 0, stream>>>(
            adj, NNODE, xc, DHID, 0, /*Cinit=*/xc, DHID, nullptr, eps, l,
            agg, DHID, NNODE, 0);
        // h = relu(agg @ W1[l]^T + b1[l])
        crgnn_gemm_wmma<<<dim3(4, 8), 128, 0, stream>>>(
            agg, DHID, W1 + l * DHID * DHID, DHID, 1, nullptr, 0,
            b1 + l * DHID, nullptr, 0, hbuf, DHID, DHID, 1);
        // x = h @ W2[l]^T + b2[l]
        crgnn_gemm_wmma<<<dim3(4, 8), 128, 0, stream>>>(
            hbuf, DHID, W2 + l * DHID * DHID, DHID, 1, nullptr, 0,
            b2 + l * DHID, nullptr, 0, xn, DHID, DHID, 0);
        float* t = xc; xc = xn; xn = t;
    }

    crgnn_colsum<<<1, DHID, 0, stream>>>(xc, out);
}